// GlobalLocalAttentionModule_42253888258443
// MI455X (gfx1250) — compile-verified
//
#include <hip/hip_runtime.h>
#include <math.h>

// ---------------- problem constants ----------------
#define Bn   2
#define Cn   64
#define Hn   40
#define Wn   40
#define HWn  1600
#define In   8
#define NHn  4
#define HDn  16
#define EPS_BN 1e-5f

typedef float v2f __attribute__((ext_vector_type(2)));
typedef float v8f __attribute__((ext_vector_type(8)));

// Full-fp32 WMMA: D(16x16) = A(16x4) * B(4x16) + C
__device__ __forceinline__ v8f wmma4(v8f c, v2f a, v2f b) {
    return __builtin_amdgcn_wmma_f32_16x16x4_f32(
        /*neg_a=*/false, a, /*neg_b=*/false, b,
        /*c_mod=*/(short)0, c, /*reuse_a=*/false, /*reuse_b=*/false);
}

// ---------------- kernel 1: depthwise 3x3 + BN1 + relu ----------------
__global__ void k_dwconv(const float* __restrict__ x, const float* __restrict__ dww,
                         const float* __restrict__ dwb,
                         const float* __restrict__ g, const float* __restrict__ bb,
                         const float* __restrict__ mm, const float* __restrict__ vv,
                         float* __restrict__ sp) {
    int idx = blockIdx.x * blockDim.x + threadIdx.x;
    if (idx >= Bn * Cn * HWn) return;
    int p = idx % HWn;
    int c = (idx / HWn) % Cn;
    int b = idx / (Cn * HWn);
    int hy = p / Wn, wx = p % Wn;
    const float* xc = x + ((size_t)b * Cn + c) * HWn;
    float acc = dwb[c];
    #pragma unroll
    for (int ky = 0; ky < 3; ++ky) {
        int yy = hy + ky - 1;
        if (yy < 0 || yy >= Hn) continue;
        #pragma unroll
        for (int kx = 0; kx < 3; ++kx) {
            int xx = wx + kx - 1;
            if (xx < 0 || xx >= Wn) continue;
            acc += xc[yy * Wn + xx] * dww[c * 9 + ky * 3 + kx];
        }
    }
    float scale = g[c] * rsqrtf(vv[c] + EPS_BN);
    float val = (acc - mm[c]) * scale + bb[c];
    sp[idx] = fmaxf(val, 0.0f);
}

// ---------------- kernel 2: channel reduce conv (M=16,K=64) + BN2 + relu ----------------
__global__ void k_conv_ch(const float* __restrict__ Wt, const float* __restrict__ bias,
                          const float* __restrict__ g, const float* __restrict__ bb,
                          const float* __restrict__ mm, const float* __restrict__ vv,
                          const float* __restrict__ X, float* __restrict__ Y) {
    int lane = threadIdx.x, half = lane >> 4, m = lane & 15;
    int nb = blockIdx.x * 16, b = blockIdx.z;
    const float* Xb = X + (size_t)b * Cn * HWn;
    v8f acc = {};
    for (int k0 = 0; k0 < Cn; k0 += 4) {
        v2f a, bf;
        const float* wp = Wt + m * Cn + k0 + 2 * half;
        a.x = wp[0]; a.y = wp[1];
        const float* xp = Xb + (size_t)(k0 + 2 * half) * HWn + nb + m;
        bf.x = xp[0]; bf.y = xp[HWn];
        acc = wmma4(acc, a, bf);
    }
    float* Yb = Y + (size_t)b * 16 * HWn;
    #pragma unroll
    for (int r = 0; r < 8; ++r) {
        int row = half * 8 + r;
        float scale = g[row] * rsqrtf(vv[row] + EPS_BN);
        float val = (acc[r] + bias[row] - mm[row]) * scale + bb[row];
        Yb[(size_t)row * HWn + nb + m] = fmaxf(val, 0.0f);
    }
}

// ---------------- generic conv1x1 (M=64) with optional residual add ----------------
__global__ void k_conv_plain(const float* __restrict__ Wt, const float* __restrict__ bias,
                             const float* __restrict__ X, const float* __restrict__ addsrc,
                             float* __restrict__ Y, int K) {
    int lane = threadIdx.x, half = lane >> 4, m = lane & 15;
    int nb = blockIdx.x * 16, mb = blockIdx.y * 16, b = blockIdx.z;
    const float* Xb = X + (size_t)b * K * HWn;
    v8f acc = {};
    for (int k0 = 0; k0 < K; k0 += 4) {
        v2f a, bf;
        const float* wp = Wt + (mb + m) * K + k0 + 2 * half;
        a.x = wp[0]; a.y = wp[1];
        const float* xp = Xb + (size_t)(k0 + 2 * half) * HWn + nb + m;
        bf.x = xp[0]; bf.y = xp[HWn];
        acc = wmma4(acc, a, bf);
    }
    float* Yb = Y + (size_t)b * Cn * HWn;
    const float* Ab = addsrc ? addsrc + (size_t)b * Cn * HWn : nullptr;
    #pragma unroll
    for (int r = 0; r < 8; ++r) {
        int row = mb + half * 8 + r;
        float val = acc[r] + bias[row];
        if (Ab) val += Ab[(size_t)row * HWn + nb + m];
        Yb[(size_t)row * HWn + nb + m] = val;
    }
}

// ---------------- pooling: per (b,c) max & mean over HW ----------------
__global__ void k_pool(const float* __restrict__ feats, float* __restrict__ mx,
                       float* __restrict__ av) {
    __shared__ float smax[256], ssum[256];
    int c = blockIdx.x, b = blockIdx.y, t = threadIdx.x;
    const float* f = feats + ((size_t)b * Cn + c) * HWn;
    float vm = -INFINITY, vs = 0.0f;
    for (int i = t; i < HWn; i += 256) { float v = f[i]; vm = fmaxf(vm, v); vs += v; }
    smax[t] = vm; ssum[t] = vs;
    __syncthreads();
    for (int s = 128; s > 0; s >>= 1) {
        if (t < s) { smax[t] = fmaxf(smax[t], smax[t + s]); ssum[t] += ssum[t + s]; }
        __syncthreads();
    }
    if (t == 0) { mx[b * Cn + c] = smax[0]; av[b * Cn + c] = ssum[0] * (1.0f / HWn); }
}

// ---------------- gating MLP: pooled[2C] -> 16 -> sigmoid gate[C] ----------------
__global__ void k_mlp(const float* __restrict__ mx, const float* __restrict__ av,
                      const float* __restrict__ w1, const float* __restrict__ b1,
                      const float* __restrict__ w2, const float* __restrict__ b2,
                      float* __restrict__ gate) {
    __shared__ float hid[Bn][16];
    int t = threadIdx.x;
    if (t < Bn * 16) {
        int b = t / 16, j = t % 16;
        float s = b1[j];
        for (int c = 0; c < Cn; ++c) {
            s += mx[b * Cn + c] * w1[j * (2 * Cn) + c];
            s += av[b * Cn + c] * w1[j * (2 * Cn) + Cn + c];
        }
        hid[b][j] = fmaxf(s, 0.0f);
    }
    __syncthreads();
    if (t < Bn * Cn) {
        int b = t / Cn, c = t % Cn;
        float s = b2[c];
        #pragma unroll
        for (int j = 0; j < 16; ++j) s += hid[b][j] * w2[c * 16 + j];
        gate[b * Cn + c] = 1.0f / (1.0f + __expf(-s));
    }
}

// ---------------- instance masks: softmax over I of inst conv ----------------
__global__ void k_inst(const float* __restrict__ x, const float* __restrict__ iw,
                       const float* __restrict__ ib, float* __restrict__ masks) {
    int idx = blockIdx.x * blockDim.x + threadIdx.x;
    if (idx >= Bn * HWn) return;
    int p = idx % HWn, b = idx / HWn;
    const float* xb = x + (size_t)b * Cn * HWn + p;
    float lg[In];
    float mmax = -INFINITY;
    #pragma unroll
    for (int i = 0; i < In; ++i) {
        float s = ib[i];
        for (int c = 0; c < Cn; ++c) s += iw[i * Cn + c] * xb[(size_t)c * HWn];
        lg[i] = s; mmax = fmaxf(mmax, s);
    }
    float denom = 0.0f;
    #pragma unroll
    for (int i = 0; i < In; ++i) { lg[i] = __expf(lg[i] - mmax); denom += lg[i]; }
    float inv = 1.0f / denom;
    #pragma unroll
    for (int i = 0; i < In; ++i) masks[((size_t)i * Bn + b) * HWn + p] = lg[i] * inv;
}

// ---------------- QKV conv with fused mask multiply, optional per-head L2 norm ----------------
__global__ void k_conv_qkv(const float* __restrict__ Wt, const float* __restrict__ bias,
                           const float* __restrict__ x, const float* __restrict__ masks,
                           float* __restrict__ out, int doNorm) {
    int lane = threadIdx.x, half = lane >> 4, m = lane & 15;
    int nb = blockIdx.x * 16, mb = blockIdx.y * 16;
    int z = blockIdx.z;            // z = i*B + b
    int b = z % Bn;
    const float* Xb = x + (size_t)b * Cn * HWn;
    float mv = masks[(size_t)z * HWn + nb + m];   // per-pixel mask, pixel == column
    v8f acc = {};
    for (int k0 = 0; k0 < Cn; k0 += 4) {
        v2f a, bf;
        const float* wp = Wt + (mb + m) * Cn + k0 + 2 * half;
        a.x = wp[0]; a.y = wp[1];
        const float* xp = Xb + (size_t)(k0 + 2 * half) * HWn + nb + m;
        bf.x = xp[0] * mv; bf.y = xp[HWn] * mv;
        acc = wmma4(acc, a, bf);
    }
    float vals[8];
    #pragma unroll
    for (int r = 0; r < 8; ++r) vals[r] = acc[r] + bias[mb + half * 8 + r];
    if (doNorm) {
        // head == this 16-row tile; L2 norm over the 16 rows for each column
        float ss = 0.0f;
        #pragma unroll
        for (int r = 0; r < 8; ++r) ss += vals[r] * vals[r];
        ss += __shfl_xor(ss, 16, 32);
        float n = fmaxf(sqrtf(ss), 1e-12f);
        float invn = 1.0f / n;
        #pragma unroll
        for (int r = 0; r < 8; ++r) vals[r] *= invn;
    }
    float* Yz = out + (size_t)z * Cn * HWn;
    #pragma unroll
    for (int r = 0; r < 8; ++r)
        Yz[(size_t)(mb + half * 8 + r) * HWn + nb + m] = vals[r];
}

// ---------------- flash attention: one wave per (b,h,qtile), loop instances ----------------
__global__ void k_flash(const float* __restrict__ qb, const float* __restrict__ kb,
                        const float* __restrict__ vb, const float* __restrict__ masks,
                        float* __restrict__ localout) {
    __shared__ float Vlds[256];
    int lane = threadIdx.x, half = lane >> 4, m = lane & 15;
    int qt = blockIdx.x, h = blockIdx.y, b = blockIdx.z;
    float total[8];
    #pragma unroll
    for (int r = 0; r < 8; ++r) total[r] = 0.0f;

    for (int i = 0; i < In; ++i) {
        int z = i * Bn + b;
        const float* qp = qb + ((size_t)z * Cn + h * HDn) * HWn + qt * 16;
        const float* kp = kb + ((size_t)z * Cn + h * HDn) * HWn;
        const float* vp = vb + ((size_t)z * Cn + h * HDn) * HWn;
        // preload Q fragments (B operand of S-WMMA): rows = channel, col = this query
        v2f qf[4];
        #pragma unroll
        for (int s = 0; s < 4; ++s) {
            int c0 = s * 4 + 2 * half;
            qf[s].x = qp[(size_t)c0 * HWn + m];
            qf[s].y = qp[(size_t)(c0 + 1) * HWn + m];
        }
        float mrun = -INFINITY, l = 0.0f;
        v8f acco = {};
        for (int kt = 0; kt < HWn / 16; ++kt) {
            __syncthreads();
            #pragma unroll
            for (int j = 0; j < 8; ++j) {               // stage V tile [c][kp] into LDS
                int idx = lane + 32 * j;
                int c = idx >> 4, kpix = idx & 15;
                Vlds[idx] = vp[(size_t)c * HWn + kt * 16 + kpix];
            }
            __syncthreads();
            // S[key,query] = K^T Q   (contraction over HD channels)
            v8f accs = {};
            const float* kpt = kp + kt * 16;
            #pragma unroll
            for (int s = 0; s < 4; ++s) {
                int c0 = s * 4 + 2 * half;
                v2f a;
                a.x = kpt[(size_t)c0 * HWn + m];
                a.y = kpt[(size_t)(c0 + 1) * HWn + m];
                accs = wmma4(accs, a, qf[s]);
            }
            // online softmax over keys (M dim): in-lane 8 + partner via shfl_xor(16)
            float sv[8], p[8], pp[8];
            float tm = -INFINITY;
            #pragma unroll
            for (int r = 0; r < 8; ++r) { sv[r] = accs[r] * 0.25f; tm = fmaxf(tm, sv[r]); }
            tm = fmaxf(tm, __shfl_xor(tm, 16, 32));
            float mn = fmaxf(mrun, tm);
            float alpha = __expf(mrun - mn);
            float ts = 0.0f;
            #pragma unroll
            for (int r = 0; r < 8; ++r) { p[r] = __expf(sv[r] - mn); ts += p[r]; }
            ts += __shfl_xor(ts, 16, 32);
            l = l * alpha + ts;
            #pragma unroll
            for (int r = 0; r < 8; ++r) acco[r] *= alpha;
            #pragma unroll
            for (int r = 0; r < 8; ++r) pp[r] = __shfl_xor(p[r], 16, 32);
            mrun = mn;
            // O[c,query] += V[c,key] * P[key,query]
            #pragma unroll
            for (int s = 0; s < 4; ++s) {
                const int k0 = s * 4;
                v2f a;
                a.x = Vlds[m * 16 + k0 + 2 * half];
                a.y = Vlds[m * 16 + k0 + 2 * half + 1];
                // B fragment rows k0+2*half, +1 for this lane's query column:
                float bx0 = (k0 < 8)     ? p[k0]      : pp[k0 - 8];
                float bx1 = (k0 + 2 < 8) ? pp[k0 + 2] : p[k0 - 6];
                float by0 = (k0 + 1 < 8) ? p[k0 + 1]  : pp[k0 - 7];
                float by1 = (k0 + 3 < 8) ? pp[k0 + 3] : p[k0 - 5];
                v2f bf;
                bf.x = half ? bx1 : bx0;
                bf.y = half ? by1 : by0;
                acco = wmma4(acco, a, bf);
            }
        }
        float mv = masks[(size_t)z * HWn + qt * 16 + m];
        float invl = 1.0f / l;
        #pragma unroll
        for (int r = 0; r < 8; ++r) total[r] += acco[r] * invl * mv;
    }
    #pragma unroll
    for (int r = 0; r < 8; ++r)
        localout[(size_t)b * Cn * HWn + (size_t)(h * HDn + half * 8 + r) * HWn + qt * 16 + m] = total[r];
}

// ---------------- fusion conv (M=64,K=128) over concat(feats*gate, local) + x ----------------
__global__ void k_fuse(const float* __restrict__ Wt, const float* __restrict__ bias,
                       const float* __restrict__ feats, const float* __restrict__ gate,
                       const float* __restrict__ localb, const float* __restrict__ x,
                       float* __restrict__ out) {
    int lane = threadIdx.x, half = lane >> 4, m = lane & 15;
    int nb = blockIdx.x * 16, mb = blockIdx.y * 16, b = blockIdx.z;
    const float* Fb = feats + (size_t)b * Cn * HWn;
    const float* Lb = localb + (size_t)b * Cn * HWn;
    const float* Gb = gate + (size_t)b * Cn;
    v8f acc = {};
    for (int k0 = 0; k0 < 2 * Cn; k0 += 4) {
        int ka = k0 + 2 * half, kb2 = ka + 1;
        v2f a, bf;
        const float* wp = Wt + (mb + m) * (2 * Cn) + ka;
        a.x = wp[0]; a.y = wp[1];
        bf.x = (ka < Cn) ? Fb[(size_t)ka * HWn + nb + m] * Gb[ka]
                         : Lb[(size_t)(ka - Cn) * HWn + nb + m];
        bf.y = (kb2 < Cn) ? Fb[(size_t)kb2 * HWn + nb + m] * Gb[kb2]
                          : Lb[(size_t)(kb2 - Cn) * HWn + nb + m];
        acc = wmma4(acc, a, bf);
    }
    const float* Xb = x + (size_t)b * Cn * HWn;
    float* Ob = out + (size_t)b * Cn * HWn;
    #pragma unroll
    for (int r = 0; r < 8; ++r) {
        int row = mb + half * 8 + r;
        Ob[(size_t)row * HWn + nb + m] =
            acc[r] + bias[row] + Xb[(size_t)row * HWn + nb + m];
    }
}

// ---------------- host side ----------------
extern "C" void kernel_launch(void* const* d_in, const int* in_sizes, int n_in,
                              void* d_out, int out_size, void* d_ws, size_t ws_size,
                              hipStream_t stream) {
    const float* x       = (const float*)d_in[0];
    const float* dw_w    = (const float*)d_in[1];
    const float* dw_b    = (const float*)d_in[2];
    const float* bn1_g   = (const float*)d_in[3];
    const float* bn1_b   = (const float*)d_in[4];
    const float* bn1_m   = (const float*)d_in[5];
    const float* bn1_v   = (const float*)d_in[6];
    const float* ch_w    = (const float*)d_in[7];
    const float* ch_b    = (const float*)d_in[8];
    const float* bn2_g   = (const float*)d_in[9];
    const float* bn2_b   = (const float*)d_in[10];
    const float* bn2_m   = (const float*)d_in[11];
    const float* bn2_v   = (const float*)d_in[12];
    const float* rs_w    = (const float*)d_in[13];
    const float* rs_b    = (const float*)d_in[14];
    const float* ga_w    = (const float*)d_in[15];
    const float* ga_b    = (const float*)d_in[16];
    const float* mlp_w1  = (const float*)d_in[17];
    const float* mlp_b1  = (const float*)d_in[18];
    const float* mlp_w2  = (const float*)d_in[19];
    const float* mlp_b2  = (const float*)d_in[20];
    const float* inst_w  = (const float*)d_in[21];
    const float* inst_b  = (const float*)d_in[22];
    const float* q_w     = (const float*)d_in[23];
    const float* q_b     = (const float*)d_in[24];
    const float* kk_w    = (const float*)d_in[25];
    const float* kk_b    = (const float*)d_in[26];
    const float* v_w     = (const float*)d_in[27];
    const float* v_b     = (const float*)d_in[28];
    const float* fuse_w  = (const float*)d_in[29];
    const float* fuse_b  = (const float*)d_in[30];
    float* out = (float*)d_out;

    // workspace carve-up (floats)
    float* ws   = (float*)d_ws;
    float* sp    = ws;                               // B*C*HW     = 204800
    float* ch    = sp    + (size_t)Bn * Cn * HWn;    // B*16*HW    = 51200
    float* tbuf  = ch    + (size_t)Bn * 16 * HWn;    // B*C*HW
    float* feats = tbuf  + (size_t)Bn * Cn * HWn;    // B*C*HW
    float* mx    = feats + (size_t)Bn * Cn * HWn;    // B*C
    float* av    = mx    + (size_t)Bn * Cn;          // B*C
    float* gate  = av    + (size_t)Bn * Cn;          // B*C
    float* masks = gate  + (size_t)Bn * Cn;          // I*B*HW
    float* qbuf  = masks + (size_t)In * Bn * HWn;    // I*B*C*HW
    float* kbuf  = qbuf  + (size_t)In * Bn * Cn * HWn;
    float* vbuf  = kbuf  + (size_t)In * Bn * Cn * HWn;
    float* lbuf  = vbuf  + (size_t)In * Bn * Cn * HWn; // B*C*HW

    const int NT = HWn / 16;   // 100 column tiles

    // 1) depthwise 3x3 + BN + relu
    k_dwconv<<<(Bn * Cn * HWn + 255) / 256, 256, 0, stream>>>(
        x, dw_w, dw_b, bn1_g, bn1_b, bn1_m, bn1_v, sp);
    // 2) channel reduce + BN + relu (WMMA)
    k_conv_ch<<<dim3(NT, 1, Bn), 32, 0, stream>>>(
        ch_w, ch_b, bn2_g, bn2_b, bn2_m, bn2_v, sp, ch);
    // 3) restore conv + residual x (WMMA)
    k_conv_plain<<<dim3(NT, 4, Bn), 32, 0, stream>>>(rs_w, rs_b, ch, x, tbuf, 16);
    // 4) ga conv (WMMA)
    k_conv_plain<<<dim3(NT, 4, Bn), 32, 0, stream>>>(ga_w, ga_b, tbuf, nullptr, feats, Cn);
    // 5) pooling
    k_pool<<<dim3(Cn, Bn), 256, 0, stream>>>(feats, mx, av);
    // 6) gate MLP
    k_mlp<<<1, 128, 0, stream>>>(mx, av, mlp_w1, mlp_b1, mlp_w2, mlp_b2, gate);
    // 7) instance masks
    k_inst<<<(Bn * HWn + 127) / 128, 128, 0, stream>>>(x, inst_w, inst_b, masks);
    // 8) masked QKV (WMMA; q/k with per-head L2 norm)
    k_conv_qkv<<<dim3(NT, 4, In * Bn), 32, 0, stream>>>(q_w,  q_b,  x, masks, qbuf, 1);
    k_conv_qkv<<<dim3(NT, 4, In * Bn), 32, 0, stream>>>(kk_w, kk_b, x, masks, kbuf, 1);
    k_conv_qkv<<<dim3(NT, 4, In * Bn), 32, 0, stream>>>(v_w,  v_b,  x, masks, vbuf, 0);
    // 9) flash attention (WMMA QK^T + PV, online softmax), sums over instances
    k_flash<<<dim3(NT, NHn, Bn), 32, 0, stream>>>(qbuf, kbuf, vbuf, masks, lbuf);
    // 10) fusion conv + residual (WMMA, gate/concat fused into B-operand load)
    k_fuse<<<dim3(NT, 4, Bn), 32, 0, stream>>>(
        fuse_w, fuse_b, feats, gate, lbuf, x, out);
}